// OnlineEmbedding_82368882802753
// MI455X (gfx1250) — compile-verified
//
#include <hip/hip_runtime.h>
#include <stdint.h>

// OnlineEmbedding gather: out[row, :] = table[ids[row], :]
// rows = 4096*200 = 819200, EMBED_DIM = 64 f32 => 256 B per row.
//
// VGPR-free DMA pipeline using CDNA5 async LDS ops, double-buffered:
//   16 lanes x 16B = one 256B row; wave32 = 2 rows per async instruction.
//   Batch = 8 stages (16 rows, 4KB). Per iteration:
//     s_wait_asynccnt 0            // cur loads landed; prev stores drained
//     8x global_store_async_from_lds_b128 (cur, NT)   // contiguous out
//     8x global_load_async_to_lds_b128  (next -> other buffer)
//   so output stores overlap the next batch's gather loads.

#define EMBED_DIM        64
#define ROW_BYTES        256
#define ROWS_PER_STAGE   2                            // 32 lanes / 16
#define STAGES           8
#define ROWS_PER_BATCH   (ROWS_PER_STAGE * STAGES)    // 16
#define WAVES_PER_BLOCK  8
#define BLOCK_THREADS    (WAVES_PER_BLOCK * 32)       // 256
#define STAGE_BYTES      512                          // 32 lanes * 16B
#define SET_BYTES        (STAGES * STAGE_BYTES)       // 4096 per buffer set
#define WAVE_LDS_BYTES   (2 * SET_BYTES)              // 8192 (double buffer)
#define BLOCK_LDS_BYTES  (WAVES_PER_BLOCK * WAVE_LDS_BYTES)  // 65536

__device__ __forceinline__ void async_gather_b128(uint32_t lds, uint64_t src) {
    asm volatile("global_load_async_to_lds_b128 %0, %1, off"
                 :: "v"(lds), "v"(src) : "memory");
}
__device__ __forceinline__ void async_store_nt_b128(uint64_t dst, uint32_t lds) {
    asm volatile("global_store_async_from_lds_b128 %0, %1, off th:TH_STORE_NT"
                 :: "v"(dst), "v"(lds) : "memory");
}
__device__ __forceinline__ void wait_async0() {
    asm volatile("s_wait_asynccnt 0x0" ::: "memory");
}

__global__ __launch_bounds__(BLOCK_THREADS)
void OnlineEmbedding_gather_async(const int* __restrict__ ids,
                                  const float* __restrict__ table,
                                  float* __restrict__ out,
                                  int nrows)
{
    __shared__ __align__(16) unsigned char smem[BLOCK_LDS_BYTES];

    const int lane       = (int)(threadIdx.x & 31u);
    const int wave       = (int)(threadIdx.x >> 5u);
    const int waveGlobal = (int)blockIdx.x * WAVES_PER_BLOCK + wave;
    const int totalWaves = (int)gridDim.x * WAVES_PER_BLOCK;

    const int rowInPair = lane >> 4;   // which of the 2 rows this lane serves
    const int seg       = lane & 15;   // 16B segment within the row

    // Low 32 bits of the flat shared-memory address == LDS byte offset.
    const uint32_t ldsWave =
        (uint32_t)(uintptr_t)(&smem[wave * WAVE_LDS_BYTES]);
    const uint32_t ldsLane = ldsWave + (uint32_t)(lane * 16);

    const uint64_t tableBase = (uint64_t)(uintptr_t)table;
    const uint64_t outBase   = (uint64_t)(uintptr_t)out;

    const int nbatch = nrows / ROWS_PER_BATCH;

    int b = waveGlobal;
    if (b < nbatch) {
        // ---- Prologue: gather loads for first batch into set 0 ----
        {
            const int row0 = b * ROWS_PER_BATCH;
            const int idv  = ids[row0 + (lane & 15)];   // 16 batch ids, duplicated
            #pragma unroll
            for (int s = 0; s < STAGES; ++s) {
                const int id = __shfl(idv, 2 * s + rowInPair, 32);
                const uint64_t src = tableBase
                    + (uint64_t)(uint32_t)id * ROW_BYTES + (uint64_t)(seg * 16);
                async_gather_b128(ldsLane + (uint32_t)(s * STAGE_BYTES), src);
            }
        }

        uint32_t cur = 0;
        for (; b < nbatch; b += totalWaves) {
            // Drains: (a) this batch's loads landed, so the current LDS set is valid,
            //         (b) previous batch's stores drained, so the other LDS set is reusable.
            wait_async0();

            // ---- Stores: current LDS set to contiguous out rows (non-temporal) ----
            const int row0 = b * ROWS_PER_BATCH;
            const uint32_t curBase = ldsLane + cur * SET_BYTES;
            #pragma unroll
            for (int s = 0; s < STAGES; ++s) {
                const int rid = row0 + 2 * s + rowInPair;
                const uint64_t dst = outBase
                    + (uint64_t)rid * ROW_BYTES + (uint64_t)(seg * 16);
                async_store_nt_b128(dst, curBase + (uint32_t)(s * STAGE_BYTES));
            }

            // ---- Gather loads for next batch into the other set ----
            const int bn = b + totalWaves;
            if (bn < nbatch) {
                const int nrow0 = bn * ROWS_PER_BATCH;
                const int idv   = ids[nrow0 + (lane & 15)];
                const uint32_t nxtBase = ldsLane + (cur ^ 1u) * SET_BYTES;
                #pragma unroll
                for (int s = 0; s < STAGES; ++s) {
                    const int id = __shfl(idv, 2 * s + rowInPair, 32);
                    const uint64_t src = tableBase
                        + (uint64_t)(uint32_t)id * ROW_BYTES + (uint64_t)(seg * 16);
                    async_gather_b128(nxtBase + (uint32_t)(s * STAGE_BYTES), src);
                }
            }
            cur ^= 1u;
        }
        wait_async0();   // drain final stores
    }

    // ---- Tail rows (nrows % 16): direct b128 copies, one wave handles it ----
    const int tailStart = nbatch * ROWS_PER_BATCH;
    if (waveGlobal == 0) {
        for (int rid = tailStart + rowInPair; rid < nrows; rid += ROWS_PER_STAGE) {
            const long long id = (long long)ids[rid];
            const float4 v = ((const float4*)(table + (size_t)id * EMBED_DIM))[seg];
            ((float4*)(out + (size_t)rid * EMBED_DIM))[seg] = v;
        }
    }
}

extern "C" void kernel_launch(void* const* d_in, const int* in_sizes, int n_in,
                              void* d_out, int out_size, void* d_ws, size_t ws_size,
                              hipStream_t stream) {
    (void)n_in; (void)out_size; (void)d_ws; (void)ws_size;
    const int*   ids   = (const int*)d_in[0];    // [4096*200] int32
    const float* table = (const float*)d_in[1];  // [1000000, 64] f32
    float*       out   = (float*)d_out;          // [4096*200, 64] f32
    const int nrows = in_sizes[0];               // 819200

    // 2048 blocks x 8 waves = 16384 waves; 51200 batches -> ~3 batches/wave.
    // 64KB LDS / block (double-buffered) -> 5 blocks / 320KB WGP.
    const int blocks = 2048;
    OnlineEmbedding_gather_async<<<blocks, BLOCK_THREADS, 0, stream>>>(
        ids, table, out, nrows);
}